// LucasKanadetracking_78202764525703
// MI455X (gfx1250) — compile-verified
//
#include <hip/hip_runtime.h>

#define IMG_H 2048
#define IMG_W 2048

typedef __attribute__((ext_vector_type(2))) float v2f;
typedef __attribute__((ext_vector_type(8))) float v8f;

// Load/store through a uniform base + 32-bit byte offset: selects the
// GLOBAL_LOAD/STORE *_SADDR form (scalar base, VGPR voffset), avoiding
// per-lane 64-bit address arithmetic.
__device__ __forceinline__ float ldb(const float* __restrict__ base,
                                     unsigned byteoff) {
  return *(const float*)((const char*)base + byteoff);
}
__device__ __forceinline__ void stb(float* __restrict__ base,
                                    unsigned byteoff, float v) {
  *(float*)((char*)base + byteoff) = v;
}

// Combine the 2x2 window values of both frames into the three derivative
// responses (cross-correlation with fx/fy/ft kernels, already averaged).
__device__ __forceinline__ void res_from(float a0, float b0, float g0, float d0,
                                         float a1, float b1, float g1, float d1,
                                         float& rx, float& ry, float& rt) {
  float rx0 = (b0 - a0) + (d0 - g0);
  float rx1 = (b1 - a1) + (d1 - g1);
  float ry0 = (g0 - a0) + (d0 - b0);
  float ry1 = (g1 - a1) + (d1 - b1);
  float s0  = (a0 + b0) + (g0 + d0);
  float s1  = (a1 + b1) + (g1 + d1);
  rx = 0.5f * (rx0 + rx1);
  ry = 0.5f * (ry0 + ry1);
  rt = 0.5f * (s1 - s0);
}

// Horizontal 3-tap sum of the 5 products for one arbitrary global row
// (used for the +/-1 halo rows).  Column byte offsets are precomputed.
__device__ __forceinline__ void halo_hsum(const float* __restrict__ f0,
                                          const float* __restrict__ f1,
                                          int grow,
                                          unsigned cB, unsigned c1B,
                                          unsigned ccB, unsigned cc1B,
                                          float cvf, int g, int lm, int lp,
                                          float Hout[5]) {
  bool rvalid = (grow >= 0) && (grow < IMG_H);
  unsigned rr = rvalid ? (unsigned)grow : 0u;
  unsigned r1 = (rr + 1u >= IMG_H) ? (IMG_H - 2u) : (rr + 1u);
  unsigned roB  = rr << 13;   // * IMG_W * 4 bytes
  unsigned r1oB = r1 << 13;
  float rx, ry, rt, ex, ey, et;
  res_from(ldb(f0, roB + cB),  ldb(f0, roB + c1B),
           ldb(f0, r1oB + cB), ldb(f0, r1oB + c1B),
           ldb(f1, roB + cB),  ldb(f1, roB + c1B),
           ldb(f1, r1oB + cB), ldb(f1, r1oB + c1B), rx, ry, rt);
  res_from(ldb(f0, roB + ccB),  ldb(f0, roB + cc1B),
           ldb(f0, r1oB + ccB), ldb(f0, r1oB + cc1B),
           ldb(f1, roB + ccB),  ldb(f1, roB + cc1B),
           ldb(f1, r1oB + ccB), ldb(f1, r1oB + cc1B), ex, ey, et);
  float po[5] = { rx * rx, rx * ry, ry * ry, rx * rt, ry * rt };
  float pev[5] = { ex * ex * cvf, ex * ey * cvf, ey * ey * cvf,
                   ex * et * cvf, ey * et * cvf };
  float rvf = rvalid ? 1.0f : 0.0f;
#pragma unroll
  for (int q = 0; q < 5; ++q) {
    float pm = __shfl(po[q], lm, 32);
    float pp = __shfl(po[q], lp, 32);
    pm = (g == 0)  ? pev[q] : pm;
    pp = (g == 15) ? pev[q] : pp;
    Hout[q] = (po[q] + pm + pp) * rvf;
  }
}

__global__ __launch_bounds__(256)
void lk_tile_kernel(const float* __restrict__ f0, const float* __restrict__ f1,
                    float* __restrict__ uo, float* __restrict__ vo) {
  const int lane = threadIdx.x & 31;
  const int half = lane >> 4;            // 0: rows v, 1: rows v+8 (D-layout)
  const int wave = threadIdx.x >> 5;
  const int tilesX = IMG_W / 16;
  const int tile = blockIdx.x * 8 + wave;
  const int ty = tile / tilesX;
  const int tx = tile - ty * tilesX;
  const int gy0  = ty * 16;
  const int g    = lane & 15;
  const int gcol = tx * 16 + g;

  // Per-lane column byte offsets (reflect pad right, halo validity).
  const unsigned c  = (unsigned)gcol;
  const unsigned c1 = (c + 1u >= IMG_W) ? (IMG_W - 2u) : (c + 1u);
  const int ce = (g == 0) ? (gcol - 1) : ((g == 15) ? (gcol + 1) : gcol);
  const bool cvalid = (ce >= 0) && (ce < IMG_W);
  const unsigned cc  = cvalid ? (unsigned)ce : 0u;
  const unsigned cc1 = (cc + 1u >= IMG_W) ? (IMG_W - 2u) : (cc + 1u);
  const unsigned cB = c << 2, c1B = c1 << 2, ccB = cc << 2, cc1B = cc1 << 2;
  const float cvf = cvalid ? 1.0f : 0.0f;
  const int lm = (lane & 16) | ((lane - 1) & 15);  // stay within 16-lane group
  const int lp = (lane & 16) | ((lane + 1) & 15);

  // ---- horizontally-summed products, 16x16 tile in WMMA D-layout --------
  // Rolling row reuse: row v and row v+1 of the 2x2 windows share a row.
  float Hq[5][8];
  const unsigned rbase = (unsigned)(gy0 + half * 8);
  unsigned roB = rbase << 13;
  float pa0 = ldb(f0, roB + cB),  pb0 = ldb(f0, roB + c1B);
  float pa1 = ldb(f1, roB + cB),  pb1 = ldb(f1, roB + c1B);
  float pe0 = ldb(f0, roB + ccB), pf0 = ldb(f0, roB + cc1B);
  float pe1 = ldb(f1, roB + ccB), pf1 = ldb(f1, roB + cc1B);
#pragma unroll
  for (int v = 0; v < 8; ++v) {
    unsigned rn = rbase + (unsigned)v + 1u;
    // rn can only exceed the image on the very last unrolled iteration
    // (bottom lane-half of the bottom tile row): guard at compile time.
    unsigned rni = (v == 7) ? ((rn >= IMG_H) ? (IMG_H - 2u) : rn) : rn;
    unsigned noB = rni << 13;
    unsigned o0 = noB + cB, o1 = noB + c1B, o2 = noB + ccB, o3 = noB + cc1B;
    float na0 = ldb(f0, o0), nb0 = ldb(f0, o1);
    float na1 = ldb(f1, o0), nb1 = ldb(f1, o1);
    float ne0 = ldb(f0, o2), nf0 = ldb(f0, o3);
    float ne1 = ldb(f1, o2), nf1 = ldb(f1, o3);
    float rx, ry, rt, ex, ey, et;
    res_from(pa0, pb0, na0, nb0, pa1, pb1, na1, nb1, rx, ry, rt);
    res_from(pe0, pf0, ne0, nf0, pe1, pf1, ne1, nf1, ex, ey, et);
    float po[5] = { rx * rx, rx * ry, ry * ry, rx * rt, ry * rt };
    float pev[5] = { ex * ex * cvf, ex * ey * cvf, ey * ey * cvf,
                     ex * et * cvf, ey * et * cvf };
#pragma unroll
    for (int q = 0; q < 5; ++q) {
      float pm = __shfl(po[q], lm, 32);
      float pp = __shfl(po[q], lp, 32);
      pm = (g == 0)  ? pev[q] : pm;
      pp = (g == 15) ? pev[q] : pp;
      Hq[q][v] = po[q] + pm + pp;
    }
    pa0 = na0; pb0 = nb0; pa1 = na1; pb1 = nb1;
    pe0 = ne0; pf0 = nf0; pe1 = ne1; pf1 = nf1;
  }
  float Htop[5], Hbot[5];                 // halo rows -1 and +16
  halo_hsum(f0, f1, gy0 - 1,  cB, c1B, ccB, cc1B, cvf, g, lm, lp, Htop);
  halo_hsum(f0, f1, gy0 + 16, cB, c1B, ccB, cc1B, cvf, g, lm, lp, Hbot);
  // Mask halo rows to the lane-half whose output rows they touch
  // (row 0 lives in lanes 0-15 of VGPR0, row 15 in lanes 16-31 of VGPR7).
  float HtopM[5], HbotM[5];
#pragma unroll
  for (int q = 0; q < 5; ++q) {
    HtopM[q] = (half == 0) ? Htop[q] : 0.0f;
    HbotM[q] = (half == 0) ? 0.0f : Hbot[q];
  }

  // ---- constant tridiagonal band T, four 16x4 chunks in A-layout --------
  v2f Aband[4];
#pragma unroll
  for (int ch = 0; ch < 4; ++ch) {
    int kb = 4 * ch + half * 2;
    int d0 = g - kb;
    int d1 = g - (kb + 1);
    Aband[ch].x = (d0 >= -1 && d0 <= 1) ? 1.0f : 0.0f;
    Aband[ch].y = (d1 >= -1 && d1 <= 1) ? 1.0f : 0.0f;
  }

  // ---- vertical 3-tap sum: S = T x H via chained V_WMMA_F32_16X16X4_F32 -
  const v8f vzero = {};   // one shared zero octet for all 5 chains
  float S[5][8];
#pragma unroll
  for (int q = 0; q < 5; ++q) {
    v8f acc = vzero;
#pragma unroll
    for (int ch = 0; ch < 4; ++ch) {
      const int li = (4 * ch) & 7;   // 0,4,0,4
      const int hh = (4 * ch) >> 3;  // 0,0,1,1 : lane-half holding the rows
      float s0 = __shfl_xor(Hq[q][li + 0], 16, 32);
      float s1 = __shfl_xor(Hq[q][li + 1], 16, 32);
      float s2 = __shfl_xor(Hq[q][li + 2], 16, 32);
      float s3 = __shfl_xor(Hq[q][li + 3], 16, 32);
      v2f B;
      if (hh == 0) {
        B.x = (half == 0) ? Hq[q][li + 0] : s2;
        B.y = (half == 0) ? Hq[q][li + 1] : s3;
      } else {
        B.x = (half == 0) ? s0 : Hq[q][li + 2];
        B.y = (half == 0) ? s1 : Hq[q][li + 3];
      }
      acc = __builtin_amdgcn_wmma_f32_16x16x4_f32(
          false, Aband[ch], false, B, (short)0, acc, false, false);
    }
#pragma unroll
    for (int v = 0; v < 8; ++v) S[q][v] = acc[v];
    S[q][0] += HtopM[q];   // halo row -1 -> output row 0
    S[q][7] += HbotM[q];   // halo row 16 -> output row 15
  }

  // ---- per-pixel 2x2 solve + edge masking, store u and v ----------------
#pragma unroll
  for (int v = 0; v < 8; ++v) {
    int gr = gy0 + v + half * 8;
    float Ixx = S[0][v], Ixy = S[1][v], Iyy = S[2][v];
    float Ixt = S[3][v], Iyt = S[4][v];
    float det = Ixx * Iyy - Ixy * Ixy;
    bool nz = (det != 0.0f);
    float sd = nz ? det : 1.0f;
    float rdet = 1.0f / sd;                  // one divide, two multiplies
    bool ok = nz && (gr > 0) && (gcol > 0);  // first row/col forced to zero
    float uu = ok ? (Iyy * Ixt - Ixy * Iyt) * rdet : 0.0f;
    float vv = ok ? (Ixx * Iyt - Ixy * Ixt) * rdet : 0.0f;
    unsigned idxB = ((unsigned)gr << 13) + cB;
    stb(uo, idxB, uu);
    stb(vo, idxB, vv);
  }
}

extern "C" void kernel_launch(void* const* d_in, const int* in_sizes, int n_in,
                              void* d_out, int out_size, void* d_ws, size_t ws_size,
                              hipStream_t stream) {
  (void)in_sizes; (void)n_in; (void)out_size; (void)d_ws; (void)ws_size;
  const unsigned HW = IMG_H * IMG_W;
  const float* f0 = (const float*)d_in[0];       // frame 0
  const float* f1 = (const float*)d_in[0] + HW;  // frame 1
  float* uo = (float*)d_out;                     // u plane
  float* vo = (float*)d_out + HW;                // v plane
  const int tiles = (IMG_H / 16) * (IMG_W / 16);  // 16384
  const int blocks = tiles / 8;                   // 8 waves/block, 1 tile/wave
  lk_tile_kernel<<<blocks, 256, 0, stream>>>(f0, f1, uo, vo);
}